// FCN_81913616269760
// MI455X (gfx1250) — compile-verified
//
#include <hip/hip_runtime.h>
#include <hip/hip_bf16.h>

// 3-layer tanh RNN, B=256, T=16384, IN=1, H=16, OUT=1 (all fp32).
// Latency-bound serial recurrence: one wave32 per 16-batch tile, hidden state
// kept transposed (H^T: [hidden=16 rows, batch=16 cols]) in WMMA C/D layout,
// weight matrices preloaded once into WMMA A layout, mat-vecs done as chains
// of 4x V_WMMA_F32_16X16X4_F32. State C/D -> B-operand conversion uses
// v_permlanex16_b32 half-swaps (VALU, no LDS round-trip) + cndmask merges.

#define TT   16384
#define BB   256
#define HH   16

typedef __attribute__((ext_vector_type(2))) float v2f;
typedef __attribute__((ext_vector_type(8))) float v8f;

__device__ __forceinline__ float swap16(float v) {
  int i = __float_as_int(v);
#if __has_builtin(__builtin_amdgcn_permlanex16)
  // identity selectors -> lane i reads lane i of the opposite 16-lane row,
  // i.e. lane ^ 16 swap, executed in the VALU pipe (no DScnt wait).
  i = __builtin_amdgcn_permlanex16(i, i, 0x76543210, 0xfedcba98, false, false);
#else
  // group-of-32 swizzle: xor=0x10, or=0, and=0x1f  -> SWAPX16 (lane ^ 16)
  i = __builtin_amdgcn_ds_swizzle(i, 0x401F);
#endif
  return __int_as_float(i);
}

__device__ __forceinline__ float fast_tanh(float x) {
#if __has_builtin(__builtin_amdgcn_tanhf)
  return __builtin_amdgcn_tanhf(x);   // v_tanh_f32
#else
  return tanhf(x);
#endif
}

struct A8 { v2f a[4]; };   // 16x16 fp32 matrix in WMMA A layout, K-sliced by 4

// A layout for V_WMMA_F32_16X16X4_F32 (16x4): lane m(0-15): V0=K0,V1=K1;
// lanes 16-31: V0=K2,V1=K3.  kb = (lane<16) ? 0 : 2.
__device__ __forceinline__ A8 load_A(const float* __restrict__ W, int m, int kb) {
  A8 r;
#pragma unroll
  for (int j = 0; j < 4; ++j) {
    r.a[j].x = W[m * HH + 4 * j + kb + 0];
    r.a[j].y = W[m * HH + 4 * j + kb + 1];
  }
  return r;
}

// acc += A(16x16) * s(16x16), s given as H^T in C/D layout:
//   (n,b): n<8 -> vgpr s[n], lanes 0-15 (col=lane); n>=8 -> s[n-8], lanes 16-31.
// B_j operand (4x16, rows k=4j..4j+3): V0 = {lo:k=4j, hi:k=4j+2}, V1 = {lo:4j+1, hi:4j+3}.
__device__ __forceinline__ v8f mm16(const A8& A, const v8f& s, v8f acc, bool lo) {
  float sw[8];
#pragma unroll
  for (int i = 0; i < 8; ++i) sw[i] = swap16(s[i]);
  v2f b[4];
  b[0].x = lo ? s[0] : sw[2];  b[0].y = lo ? s[1] : sw[3];
  b[1].x = lo ? s[4] : sw[6];  b[1].y = lo ? s[5] : sw[7];
  b[2].x = lo ? sw[0] : s[2];  b[2].y = lo ? sw[1] : s[3];
  b[3].x = lo ? sw[4] : s[6];  b[3].y = lo ? sw[5] : s[7];
#pragma unroll
  for (int j = 0; j < 4; ++j)
    acc = __builtin_amdgcn_wmma_f32_16x16x4_f32(false, A.a[j], false, b[j],
                                                (short)0, acc, false, false);
  return acc;
}

__device__ __forceinline__ v8f tanh8(v8f v) {
  v8f r;
#pragma unroll
  for (int i = 0; i < 8; ++i) r[i] = fast_tanh(v[i]);
  return r;
}

__global__ __launch_bounds__(32) void rnn3_kernel(
    const float* __restrict__ x,       const float* __restrict__ prev_h0,
    const float* __restrict__ post_h0, const float* __restrict__ W_ih0,
    const float* __restrict__ W_hh0,   const float* __restrict__ b_ih0,
    const float* __restrict__ b_hh0,   const float* __restrict__ W_ih1,
    const float* __restrict__ W_hh1,   const float* __restrict__ b_ih1,
    const float* __restrict__ b_hh1,   const float* __restrict__ W_ihp,
    const float* __restrict__ W_hhp,   const float* __restrict__ b_ihp,
    const float* __restrict__ b_hhp,   float* __restrict__ y) {
  const int  l  = threadIdx.x;          // 0..31
  const bool lo = (l & 16) == 0;
  const int  m  = l & 15;
  const int  kb = lo ? 0 : 2;
  const int  b  = blockIdx.x * 16 + m;  // this lane's batch column

  // Weight matrices in WMMA A layout (one-time per-lane loads).
  const A8 Ahh0 = load_A(W_hh0, m, kb);
  const A8 Aih1 = load_A(W_ih1, m, kb);
  const A8 Ahh1 = load_A(W_hh1, m, kb);

  // Per-lane "row" constants for the C/D layout: vgpr r holds row n = r + (lo?0:8).
  v8f s0, s1, bias0, bias1, wih0v, wpv;
#pragma unroll
  for (int r = 0; r < 8; ++r) {
    const int n = r + (lo ? 0 : 8);
    s0[r]    = prev_h0[n];            // h0 broadcast over batch
    s1[r]    = prev_h0[HH + n];
    bias0[r] = b_ih0[n] + b_hh0[n];
    bias1[r] = b_ih1[n] + b_hh1[n];
    wih0v[r] = W_ih0[n];              // [H,1]
    wpv[r]   = W_ihp[n];              // [1,H]
  }
  const float whhp = W_hhp[0];
  const float bp   = b_ihp[0] + b_hhp[0];
  float       o    = post_h0[0];

  const float* __restrict__ xrow = x + (size_t)b * TT;
  float* __restrict__ yrow = y + (size_t)b * TT;

  float xv_next = xrow[0];            // software-pipelined input stream
  for (int t = 0; t < TT; ++t) {
    const float xv = xv_next;
    xv_next = xrow[(t + 1) & (TT - 1)];          // power-of-2 wrap, stays in-bounds
    const float xn = (xv + 1.0f) * 0.5f;         // min-max normalize [-1,1]->[0,1]

    // Layer-1 recurrent matmul first: independent of fresh layer-0 output,
    // lets the scheduler overlap its WMMA chain with layer 0.
    v8f c1 = mm16(Ahh1, s1, bias1, lo);

    // Layer 0: pre0 = W_ih0*xn + bias0 + W_hh0 @ h0
    v8f c0;
#pragma unroll
    for (int r = 0; r < 8; ++r) c0[r] = fmaf(wih0v[r], xn, bias0[r]);
    s0 = tanh8(mm16(Ahh0, s0, c0, lo));

    // Layer 1: pre1 = W_ih1 @ y0 + bias1 + W_hh1 @ h1
    s1 = tanh8(mm16(Aih1, s0, c1, lo));

    // Head (OUT=1): o = tanh(W_ihp . h1 + W_hhp*o + bias). Per-lane partial
    // covers rows n=r+(lo?0:8); cross-half swap completes the 16-dot.
    float p = 0.0f;
#pragma unroll
    for (int r = 0; r < 8; ++r) p = fmaf(wpv[r], s1[r], p);
    p += swap16(p);
    o = fast_tanh(fmaf(whhp, o, p + bp));

    if (lo) yrow[t] = o;              // lanes 16-31 hold duplicate columns
  }
}

extern "C" void kernel_launch(void* const* d_in, const int* in_sizes, int n_in,
                              void* d_out, int out_size, void* d_ws, size_t ws_size,
                              hipStream_t stream) {
  (void)in_sizes; (void)n_in; (void)out_size; (void)d_ws; (void)ws_size;
  const float* x       = (const float*)d_in[0];
  const float* prev_h0 = (const float*)d_in[1];
  const float* post_h0 = (const float*)d_in[2];
  const float* W_ih0   = (const float*)d_in[3];
  const float* W_hh0   = (const float*)d_in[4];
  const float* b_ih0   = (const float*)d_in[5];
  const float* b_hh0   = (const float*)d_in[6];
  const float* W_ih1   = (const float*)d_in[7];
  const float* W_hh1   = (const float*)d_in[8];
  const float* b_ih1   = (const float*)d_in[9];
  const float* b_hh1   = (const float*)d_in[10];
  const float* W_ihp   = (const float*)d_in[11];
  const float* W_hhp   = (const float*)d_in[12];
  const float* b_ihp   = (const float*)d_in[13];
  const float* b_hhp   = (const float*)d_in[14];
  float* yout          = (float*)d_out;

  rnn3_kernel<<<dim3(BB / 16), dim3(32), 0, stream>>>(
      x, prev_h0, post_h0, W_ih0, W_hh0, b_ih0, b_hh0,
      W_ih1, W_hh1, b_ih1, b_hh1, W_ihp, W_hhp, b_ihp, b_hhp, yout);
}